// HyenaOperator_2104533975836
// MI455X (gfx1250) — compile-verified
//
#include <hip/hip_runtime.h>
#include <hip/hip_bf16.h>
#include <math.h>

#define B_    8
#define L_    4096
#define D_    512
#define EMB_  16
#define HID_  64
#define NFFT  8192
#define LOG2N 13
#define NTW   (NFFT / 2)   // 4096 twiddles, 32 KB

typedef __attribute__((ext_vector_type(2))) float v2f;
typedef __attribute__((ext_vector_type(8))) float v8f;

// ---------------------------------------------------------------------------
// GEMM via V_WMMA_F32_16X16X4_F32.  C[M,N] = A[M,K] @ W[K,N] + bias[N]
// One wave -> 64M x 16N strip (4 WMMA tiles sharing one B fragment):
//   per k-step: 4x global_load_b64 (A) + 2x b32 (B) feed 4 WMMAs.
// Block = 8 waves -> 64M x 128N macro-tile.
// A 16x4 layout: lanes 0-15 hold (K=0,K=1), lanes 16-31 hold (K=2,K=3).
// C/D layout: VGPR v, lanes 0-15 -> M=v, N=lane; lanes 16-31 -> M=v+8.
// ---------------------------------------------------------------------------
__global__ __launch_bounds__(256) void gemm_wmma_f32(
    const float* __restrict__ A, const float* __restrict__ W,
    const float* __restrict__ bias, float* __restrict__ C,
    int M, int N, int K)
{
  const int wave = threadIdx.x >> 5;
  const int lane = threadIdx.x & 31;
  const int half = lane >> 4;        // 0: K pair {0,1}; 1: K pair {2,3}
  const int l16  = lane & 15;
  const int tm0  = blockIdx.y * 64;
  const int tn   = blockIdx.x * 128 + wave * 16;

  const float* arow0 = A + (size_t)(tm0 +  0 + l16) * K + half * 2;
  const float* arow1 = A + (size_t)(tm0 + 16 + l16) * K + half * 2;
  const float* arow2 = A + (size_t)(tm0 + 32 + l16) * K + half * 2;
  const float* arow3 = A + (size_t)(tm0 + 48 + l16) * K + half * 2;
  const float* bcol  = W + (size_t)(half * 2) * N + tn + l16;

  v8f acc0 = {}, acc1 = {}, acc2 = {}, acc3 = {};
  for (int k = 0; k < K; k += 4) {
    v2f b;
    b.x = bcol[0];  b.y = bcol[N];            // W[k+2h, n], W[k+2h+1, n]
    v2f a0, a1, a2, a3;
    a0.x = arow0[0]; a0.y = arow0[1];
    a1.x = arow1[0]; a1.y = arow1[1];
    a2.x = arow2[0]; a2.y = arow2[1];
    a3.x = arow3[0]; a3.y = arow3[1];
    acc0 = __builtin_amdgcn_wmma_f32_16x16x4_f32(false, a0, false, b, (short)0, acc0, false, false);
    acc1 = __builtin_amdgcn_wmma_f32_16x16x4_f32(false, a1, false, b, (short)0, acc1, false, false);
    acc2 = __builtin_amdgcn_wmma_f32_16x16x4_f32(false, a2, false, b, (short)0, acc2, false, false);
    acc3 = __builtin_amdgcn_wmma_f32_16x16x4_f32(false, a3, false, b, (short)0, acc3, false, false);
    arow0 += 4; arow1 += 4; arow2 += 4; arow3 += 4;
    bcol  += (size_t)4 * N;
  }

  const int   col = tn + l16;
  const float bv  = bias[col];
  {
    float* crow = C + (size_t)(tm0 +  0 + half * 8) * N + col;
#pragma unroll
    for (int v = 0; v < 8; ++v) crow[(size_t)v * N] = acc0[v] + bv;
  }
  {
    float* crow = C + (size_t)(tm0 + 16 + half * 8) * N + col;
#pragma unroll
    for (int v = 0; v < 8; ++v) crow[(size_t)v * N] = acc1[v] + bv;
  }
  {
    float* crow = C + (size_t)(tm0 + 32 + half * 8) * N + col;
#pragma unroll
    for (int v = 0; v < 8; ++v) crow[(size_t)v * N] = acc2[v] + bv;
  }
  {
    float* crow = C + (size_t)(tm0 + 48 + half * 8) * N + col;
#pragma unroll
    for (int v = 0; v < 8; ++v) crow[(size_t)v * N] = acc3[v] + bv;
  }
}

// ---------------------------------------------------------------------------
// Twiddle table: tw[t] = exp(-i*pi*t/NTW), t in [0, NTW)
// ---------------------------------------------------------------------------
__global__ __launch_bounds__(256) void make_twiddles(float2* __restrict__ tw)
{
  int t = blockIdx.x * blockDim.x + threadIdx.x;
  if (t < NTW) {
    float s, c;
    __sincosf(-3.14159265358979323846f * (float)t / (float)NTW, &s, &c);
    tw[t] = make_float2(c, s);
  }
}

// ---------------------------------------------------------------------------
// Filter MLP: hT[d, l] = (gelu(pe[l] @ w1 + b1) @ w2 + b2)[d]
// ---------------------------------------------------------------------------
__global__ __launch_bounds__(64) void filter_mlp(
    const float* __restrict__ w1, const float* __restrict__ b1,
    const float* __restrict__ w2, const float* __restrict__ b2,
    float* __restrict__ hT)
{
  __shared__ float hid[HID_];
  const int l = blockIdx.x;
  const int j = threadIdx.x;

  float acc = b1[j];
  const float c0 = -logf(10000.f) / (float)EMB_;
#pragma unroll
  for (int i = 0; i < EMB_ / 2; ++i) {
    float div = __expf((float)(2 * i) * c0);
    float s, c;
    __sincosf((float)l * div, &s, &c);
    acc += s * w1[(2 * i) * HID_ + j] + c * w1[(2 * i + 1) * HID_ + j];
  }
  // exact GELU: 0.5*x*(1+erf(x/sqrt(2)))
  hid[j] = 0.5f * acc * (1.f + erff(acc * 0.70710678118654752f));
  __syncthreads();

  for (int d = j; d < D_; d += HID_) {
    float o = b2[d];
#pragma unroll 8
    for (int jj = 0; jj < HID_; ++jj) o += hid[jj] * w2[jj * D_ + d];
    hT[(size_t)d * L_ + l] = o;
  }
}

// ---------------------------------------------------------------------------
// In-LDS radix-2 FFT over NFFT complex points; twiddles from LDS table.
//   fft_dif: natural-order in  -> bit-reversed out (forward; table sign)
//   fft_dit: bit-reversed in   -> natural-order out (sgn=-1 conjugates = inverse)
// Twiddle: exp(-sgn*i*pi*pos/mmax) = tw[pos << (12-stage)], conjugated if sgn<0
// ---------------------------------------------------------------------------
__device__ __forceinline__ void fft_dif(float2* data, const float2* tw,
                                        int tid, int nt, float sgn)
{
  for (int stage = LOG2N - 1; stage >= 0; --stage) {
    const int mmax = 1 << stage;
    for (int j = tid; j < NFFT / 2; j += nt) {
      const int grp = j >> stage;
      const int pos = j & (mmax - 1);
      const int i0  = (grp << (stage + 1)) + pos;
      const int i1  = i0 + mmax;
      float2 w = tw[pos << (LOG2N - 1 - stage)];
      const float ws = w.y * sgn;
      float2 u = data[i0], v = data[i1];
      data[i0] = make_float2(u.x + v.x, u.y + v.y);
      float tx = u.x - v.x, ty = u.y - v.y;
      data[i1] = make_float2(tx * w.x - ty * ws, tx * ws + ty * w.x);
    }
    __syncthreads();
  }
}

__device__ __forceinline__ void fft_dit(float2* data, const float2* tw,
                                        int tid, int nt, float sgn)
{
  for (int stage = 0; stage < LOG2N; ++stage) {
    const int mmax = 1 << stage;
    for (int j = tid; j < NFFT / 2; j += nt) {
      const int grp = j >> stage;
      const int pos = j & (mmax - 1);
      const int i0  = (grp << (stage + 1)) + pos;
      const int i1  = i0 + mmax;
      float2 w = tw[pos << (LOG2N - 1 - stage)];
      const float ws = w.y * sgn;
      float2 v = data[i1];
      float2 t = make_float2(v.x * w.x - v.y * ws, v.x * ws + v.y * w.x);
      float2 u = data[i0];
      data[i0] = make_float2(u.x + t.x, u.y + t.y);
      data[i1] = make_float2(u.x - t.x, u.y - t.y);
    }
    __syncthreads();
  }
}

// ---------------------------------------------------------------------------
// Filter forward FFT: one block per channel d; Hf[d] stored bit-reversed.
// ---------------------------------------------------------------------------
__global__ __launch_bounds__(1024) void filter_fft(
    const float* __restrict__ hT, const float2* __restrict__ twg,
    float2* __restrict__ Hf)
{
  __shared__ float2 buf[NFFT];   // 64 KB
  __shared__ float2 twl[NTW];    // 32 KB  (96 KB total of 320 KB WGP LDS)
  const int d = blockIdx.x;
  const float* src = hT + (size_t)d * L_;
  for (int i = threadIdx.x; i < NFFT; i += blockDim.x)
    buf[i] = make_float2(i < L_ ? src[i] : 0.f, 0.f);
  for (int i = threadIdx.x; i < NTW; i += blockDim.x)
    twl[i] = twg[i];
  __syncthreads();
  fft_dif(buf, twl, threadIdx.x, blockDim.x, 1.f);
  float2* dst = Hf + (size_t)d * NFFT;
  for (int i = threadIdx.x; i < NFFT; i += blockDim.x) dst[i] = buf[i];
}

// ---------------------------------------------------------------------------
// Long convolution, one block per (b, d) sequence.  Fuses:
//   load x0 from proj + 3-tap short conv  ->  LDS
//   forward FFT -> pointwise * Hf[d] -> inverse FFT
//   gate with x1*x2 (read from proj)  ->  g
// ---------------------------------------------------------------------------
__global__ __launch_bounds__(1024) void hyena_conv(
    const float* __restrict__ proj, const float2* __restrict__ Hf,
    const float2* __restrict__ twg,
    const float* __restrict__ short_w, const float* __restrict__ short_b,
    float* __restrict__ g)
{
  __shared__ float2 buf[NFFT];   // 64 KB
  __shared__ float2 twl[NTW];    // 32 KB
  const int seq = blockIdx.x;
  const int bb  = seq / D_;
  const int d   = seq % D_;
  const float w0 = short_w[d * 3 + 0];
  const float w1 = short_w[d * 3 + 1];
  const float w2 = short_w[d * 3 + 2];
  const float sb = short_b[d];

  const float* p0 = proj + (size_t)bb * L_ * (3 * D_) + d;  // x0 column d

  for (int i = threadIdx.x; i < NFFT; i += blockDim.x) {
    float v = 0.f;
    if (i < L_) {
      float xm = (i > 0)      ? p0[(size_t)(i - 1) * (3 * D_)] : 0.f;
      float xc =                p0[(size_t)i       * (3 * D_)];
      float xp = (i < L_ - 1) ? p0[(size_t)(i + 1) * (3 * D_)] : 0.f;
      v = xm * w0 + xc * w1 + xp * w2 + sb;
    }
    buf[i] = make_float2(v, 0.f);
  }
  for (int i = threadIdx.x; i < NTW; i += blockDim.x)
    twl[i] = twg[i];
  __syncthreads();

  fft_dif(buf, twl, threadIdx.x, blockDim.x, 1.f);

  const float2* Hd = Hf + (size_t)d * NFFT;
  for (int i = threadIdx.x; i < NFFT; i += blockDim.x) {
    float2 a = buf[i], h2 = Hd[i];
    buf[i] = make_float2(a.x * h2.x - a.y * h2.y,
                         a.x * h2.y + a.y * h2.x);
  }
  __syncthreads();

  fft_dit(buf, twl, threadIdx.x, blockDim.x, -1.f);  // conjugate table = inverse

  const float inv = 1.f / (float)NFFT;
  for (int i = threadIdx.x; i < L_; i += blockDim.x) {
    float x1 = p0[(size_t)i * (3 * D_) + D_];
    float x2 = p0[(size_t)i * (3 * D_) + 2 * D_];
    g[((size_t)bb * L_ + i) * D_ + d] = buf[i].x * inv * x1 * x2;
  }
}

// ---------------------------------------------------------------------------
extern "C" void kernel_launch(void* const* d_in, const int* in_sizes, int n_in,
                              void* d_out, int out_size, void* d_ws, size_t ws_size,
                              hipStream_t stream)
{
  const float* x       = (const float*)d_in[0];
  const float* in_w    = (const float*)d_in[1];
  const float* in_b    = (const float*)d_in[2];
  const float* short_w = (const float*)d_in[3];
  const float* short_b = (const float*)d_in[4];
  const float* mlp_w1  = (const float*)d_in[5];
  const float* mlp_b1  = (const float*)d_in[6];
  const float* mlp_w2  = (const float*)d_in[7];
  const float* mlp_b2  = (const float*)d_in[8];
  const float* out_w   = (const float*)d_in[9];
  const float* out_b   = (const float*)d_in[10];
  float* out = (float*)d_out;

  char* ws = (char*)d_ws;
  // Workspace layout (bytes):
  //   proj : 32768*1536*4 = 201326592   @ 0
  //   hT   :   512*4096*4 =   8388608   @ 201326592
  //   Hf   :   512*8192*8 =  33554432   @ 209715200
  //   g    : 32768* 512*4 =  67108864   @ 243269632
  //   tw   :    4096*   8 =     32768   @ 310378496   (total ~296 MB)
  float*  proj = (float*)(ws);
  float*  hT   = (float*)(ws + 201326592u);
  float2* Hf   = (float2*)(ws + 209715200u);
  float*  g    = (float*)(ws + 243269632u);
  float2* twg  = (float2*)(ws + 310378496u);

  const int M = B_ * L_;  // 32768

  // 0. twiddle table
  make_twiddles<<<dim3(NTW / 256), dim3(256), 0, stream>>>(twg);

  // 1. in_proj GEMM: proj = x @ in_w + in_b        (M x 1536)
  gemm_wmma_f32<<<dim3((3 * D_) / 128, M / 64), dim3(256), 0, stream>>>(
      x, in_w, in_b, proj, M, 3 * D_, D_);

  // 2. implicit filter MLP: hT[d, l]
  filter_mlp<<<dim3(L_), dim3(HID_), 0, stream>>>(
      mlp_w1, mlp_b1, mlp_w2, mlp_b2, hT);

  // 3. filter spectra (bit-reversed order)
  filter_fft<<<dim3(D_), dim3(1024), 0, stream>>>(hT, twg, Hf);

  // 4. fused shortconv + FFT conv + gating
  hyena_conv<<<dim3(B_ * D_), dim3(1024), 0, stream>>>(
      proj, Hf, twg, short_w, short_b, g);

  // 5. out_proj GEMM: out = g @ out_w + out_b      (M x 512)
  gemm_wmma_f32<<<dim3(D_ / 128, M / 64), dim3(256), 0, stream>>>(
      g, out_w, out_b, out, M, D_, D_);
}